// MatrixVectorScaledDotProductAttention_41583873359978
// MI455X (gfx1250) — compile-verified
//
#include <hip/hip_runtime.h>

typedef __attribute__((ext_vector_type(2))) float v2f;
typedef __attribute__((ext_vector_type(8))) float v8f;

#define NB 1024
#define L  2048
#define DK 128
#define DV 128
#define NWAVES 8
#define BLOCK (NWAVES * 32)

__launch_bounds__(BLOCK)
__global__ void attn_wmma_kernel(const float* __restrict__ q,
                                 const float* __restrict__ k,
                                 const float* __restrict__ v,
                                 float* __restrict__ out,       // [NB, DV]
                                 float* __restrict__ attn_out)  // [NB, L]
{
    __shared__ float s_scores[L];     // raw scores, then normalized attn
    __shared__ float s_red[BLOCK];

    const int b      = blockIdx.x;
    const int tid    = threadIdx.x;
    const int wave   = tid >> 5;
    const int lane   = tid & 31;
    const int half   = lane >> 4;   // 0: lanes 0-15, 1: lanes 16-31
    const int lane16 = lane & 15;

    const float* __restrict__ qb = q + (size_t)b * DK;
    const float* __restrict__ kb = k + (size_t)b * L * DK;
    const float* __restrict__ vb = v + (size_t)b * L * DV;

    // ---------------- Stage 1: scores[l] = <K[l,:], q> via WMMA f32 16x16x4
    // B fragment (4x16, q broadcast to all 16 columns), preloaded for all 32 K-steps.
    // f32 B layout: lane holds column N=lane%16; lanes 0-15 K={0,1}, lanes 16-31 K={2,3}.
    v2f bq[32];
#pragma unroll
    for (int i = 0; i < 32; ++i) {
        const int d = 4 * i + 2 * half;
        bq[i].x = qb[d];
        bq[i].y = qb[d + 1];
    }

    // Each wave covers 256 rows of K in 16-row tiles.
    for (int chunk = 0; chunk < 256; chunk += 16) {
        const int row0 = wave * 256 + chunk;
        v8f c = {0.f, 0.f, 0.f, 0.f, 0.f, 0.f, 0.f, 0.f};
        // f32 A layout (16x4): row M=lane%16; lanes 0-15 K={0,1}, lanes 16-31 K={2,3}.
        const float* __restrict__ arow =
            kb + (size_t)(row0 + lane16) * DK + 2 * half;
#pragma unroll
        for (int i = 0; i < 32; ++i) {
            v2f a;
            a.x = arow[4 * i];
            a.y = arow[4 * i + 1];
            c = __builtin_amdgcn_wmma_f32_16x16x4_f32(
                    false, a, false, bq[i], (short)0, c, false, false);
        }
        // D: every column identical; VGPR j holds score(row0+j) on lanes 0-15
        // and score(row0+8+j) on lanes 16-31. Lanes 0 and 16 write 8 each.
        if (lane16 == 0) {
            const int base = row0 + 8 * half;
#pragma unroll
            for (int j = 0; j < 8; ++j) s_scores[base + j] = c[j];
        }
    }
    __syncthreads();

    // ---------------- Softmax over L (scale applied after max-subtraction)
    const float scale = 0.08838834764831845f;  // 1/sqrt(128)

    float m = -1e30f;
#pragma unroll
    for (int j = 0; j < L / BLOCK; ++j)
        m = fmaxf(m, s_scores[tid + j * BLOCK]);
    s_red[tid] = m;
    __syncthreads();
    for (int s = BLOCK / 2; s > 0; s >>= 1) {
        if (tid < s) s_red[tid] = fmaxf(s_red[tid], s_red[tid + s]);
        __syncthreads();
    }
    m = s_red[0];
    __syncthreads();

    float e[L / BLOCK];
    float lsum = 0.f;
#pragma unroll
    for (int j = 0; j < L / BLOCK; ++j) {
        e[j] = __expf((s_scores[tid + j * BLOCK] - m) * scale);
        lsum += e[j];
    }
    s_red[tid] = lsum;
    __syncthreads();
    for (int s = BLOCK / 2; s > 0; s >>= 1) {
        if (tid < s) s_red[tid] += s_red[tid + s];
        __syncthreads();
    }
    const float inv = 1.0f / s_red[0];

    // normalized attn -> LDS (for stage 2) and global output (coalesced)
#pragma unroll
    for (int j = 0; j < L / BLOCK; ++j) {
        const float a = e[j] * inv;
        s_scores[tid + j * BLOCK] = a;
        attn_out[(size_t)b * L + tid + j * BLOCK] = a;
    }
    __syncthreads();

    // ---------------- Stage 2: out[d] = sum_l attn[l] * V[l][d] via WMMA
    // Wave w owns d-tile [16w, 16w+16). A = attn broadcast rows, B = V tile.
    {
        const int d0 = wave * 16;
        v8f c = {0.f, 0.f, 0.f, 0.f, 0.f, 0.f, 0.f, 0.f};
        const float* __restrict__ vcol = vb + d0 + lane16;
#pragma unroll 4
        for (int l0 = 0; l0 < L; l0 += 4) {
            const int lk = l0 + 2 * half;
            v2f a;                       // A[m][k] = attn[l0+k] for all rows m
            a.x = s_scores[lk];
            a.y = s_scores[lk + 1];
            v2f bb;                      // B[k][n] = V[l0+k][d0+n]
            bb.x = vcol[(size_t)lk * DV];
            bb.y = vcol[(size_t)(lk + 1) * DV];
            c = __builtin_amdgcn_wmma_f32_16x16x4_f32(
                    false, a, false, bb, (short)0, c, false, false);
        }
        // D: all rows identical -> c[0] on lane n (0-15) = out[d0+n]
        if (half == 0)
            out[(size_t)b * DV + d0 + lane16] = c[0];
    }
}

extern "C" void kernel_launch(void* const* d_in, const int* in_sizes, int n_in,
                              void* d_out, int out_size, void* d_ws, size_t ws_size,
                              hipStream_t stream) {
    const float* q = (const float*)d_in[0];
    const float* k = (const float*)d_in[1];
    const float* v = (const float*)d_in[2];
    float* out  = (float*)d_out;                       // output: NB*DV floats
    float* attn = (float*)d_out + (size_t)NB * DV;     // then attn: NB*L floats
    attn_wmma_kernel<<<NB, BLOCK, 0, stream>>>(q, k, v, out, attn);
}